// SmoothnessPrior_16312285790740
// MI455X (gfx1250) — compile-verified
//
#include <hip/hip_runtime.h>

// ---------------------------------------------------------------------------
// SmoothnessPrior on MI455X (gfx1250, wave32)
//   out = -sum(flux * conv2d_same(flux, gauss17x17))
// Memory-bound (64 MiB flux @ 23.3 TB/s ~ 3 us) -> single fused pass:
//   * Tensor Data Mover DMAs each 80x80 halo tile global->LDS (no VGPR
//     staging), with TDM padding (1 DWORD per 16) giving a conflict-free
//     85-float row stride. Edge tiles take a zero-padded scalar path.
//   * Gaussian kernel is rank-1 => separable 17-tap filters, each a
//     banded-Toeplitz matmul on V_WMMA_F32_16X16X4_F32 (f32 throughout:
//     the result is a 16.7M-element dot product).
//   * Scalar result via global_atomic_add_f32.
// ---------------------------------------------------------------------------

typedef float        v2f __attribute__((ext_vector_type(2)));
typedef float        v8f __attribute__((ext_vector_type(8)));
typedef unsigned int v4u __attribute__((ext_vector_type(4)));
typedef int          v4i __attribute__((ext_vector_type(4)));
typedef int          v8i __attribute__((ext_vector_type(8)));

#define IMG_H   2048
#define IMG_W   2048
#define KS      17
#define HALO    8              // (KS-1)/2
#define TS      64             // output tile
#define ITILE   (TS + 2*HALO)  // 80: input halo tile edge
#define ISTR    85             // LDS row stride in floats (TDM pad: +1 DWORD
                               // per 16 -> 80 + 80/16; 21*lane mod 64 banks
                               // are all distinct for the 16 fragment rows)
#define MSTR    68             // LDS stride for mid tile
#define NTHREADS 256           // 8 wave32s

#if defined(__has_builtin)
#if __has_builtin(__builtin_amdgcn_tensor_load_to_lds) && \
    __has_builtin(__builtin_amdgcn_s_wait_tensorcnt)
#define USE_TDM 1
#endif
#endif

// element (jr, jc) of the halo tile -> LDS float offset (TDM pad layout)
__device__ __forceinline__ int tileOff(int jr, int jc) {
    return jr * ISTR + jc + (jc >> 4);
}

// ---- prep: factorize 17x17 kernel into row/col taps, zero the output -----
// Rank-1 identity (kernel sums to 1): K[i][j] == rowsum[i] * colsum[j].
__global__ void sp_prep_kernel(const float* __restrict__ k2d,
                               float* __restrict__ taps,
                               float* __restrict__ out) {
    int t = threadIdx.x;
    if (t < KS) {                       // column sums -> wx
        float s = 0.f;
        for (int i = 0; i < KS; ++i) s += k2d[i * KS + t];
        taps[t] = s;
    } else if (t < 2 * KS) {            // row sums -> wy
        int r = t - KS;
        float s = 0.f;
        for (int j = 0; j < KS; ++j) s += k2d[r * KS + j];
        taps[KS + r] = s;
    }
    if (t == 63) out[0] = 0.f;          // harness poisons d_out; re-zero
}

// ---- main fused kernel ----------------------------------------------------
__global__ void __launch_bounds__(NTHREADS)
sp_main_kernel(const float* __restrict__ flux,
               const float* __restrict__ taps,
               float* __restrict__ out) {
    __shared__ float s_in [ITILE * ISTR];   // input halo tile  (27.2 KB)
    __shared__ float s_mid[ITILE * MSTR];   // after x-conv     (21.8 KB)
    __shared__ float s_wx[48];              // wx padded: W[d], d in [-15,31]
    __shared__ float s_wy[48];
    __shared__ float s_red[8];

    const int tid  = threadIdx.x;
    const int lane = tid & 31;
    const int wv   = tid >> 5;              // wave id 0..7
    const int cc   = lane & 15;             // N-column / M-row within frag
    const int hs   = (lane >> 4);           // half-wave select (K +2 / M +8)

    const int    plane = IMG_H * IMG_W;
    const float* img   = flux + (size_t)blockIdx.z * plane;
    const int gx0 = blockIdx.x * TS - HALO;
    const int gy0 = blockIdx.y * TS - HALO;

    const bool interior = (gx0 >= 0) && (gy0 >= 0) &&
                          (gx0 + ITILE <= IMG_W) && (gy0 + ITILE <= IMG_H);

#ifdef USE_TDM
    if (interior) {
        // --- Tensor Data Mover: DMA 80x80 f32 tile, row stride 2048 ------
        if (wv == 0) {                       // wave-level op; one wave issues
            unsigned           lds = (unsigned)(uintptr_t)(void*)&s_in[0];
            unsigned long long ga  = (unsigned long long)(uintptr_t)
                                     (img + (size_t)gy0 * IMG_W + gx0);
            // D# group 0: count=1 | lds_addr | global_addr[56:0] | type=2
            v4u g0 = { 1u, lds, (unsigned)ga,
                       (unsigned)((ga >> 32) & 0x1FFFFFFu) | 0x80000000u };
            // D# group 1: data_size=4B(2), pad_enable, pad_interval=16DW(3),
            // pad_amount=1DW(0); tensor_dim0/1=2048; tile 80x80; stride 2048
            v8i g1 = { (int)((2u << 16) | (1u << 20) | (3u << 22)),
                       (int)(2048u << 16),      // tensor_dim0[15:0] (hi half)
                       (int)(2048u << 16),      // dim0[31:16]=0 | dim1[15:0]
                       (int)(80u << 16),        // dim1[31:16]=0 | tile_dim0
                       80,                      // tile_dim1 | tile_dim2=0
                       2048,                    // tensor_dim0_stride[31:0]
                       0, 0 };                  // stride hi | dim1_stride
            v4i gz4 = { 0, 0, 0, 0 };           // groups 2/3: unused (2D)
            v8i gz8 = { 0, 0, 0, 0, 0, 0, 0, 0 };
            __builtin_amdgcn_tensor_load_to_lds(g0, g1, gz4, gz4, gz8, 0);
            __builtin_amdgcn_s_wait_tensorcnt(0);   // TENSORcnt is per-wave
        }
    } else
#endif
    {
        // --- scalar fallback with zero-padded edges (same LDS layout) ----
        for (int i = tid; i < ITILE * ITILE; i += NTHREADS) {
            int jr = i / ITILE, jc = i - jr * ITILE;
            int gy = gy0 + jr,  gx = gx0 + jc;
            float v = 0.f;
            if ((unsigned)gy < (unsigned)IMG_H && (unsigned)gx < (unsigned)IMG_W)
                v = img[gy * IMG_W + gx];
            s_in[tileOff(jr, jc)] = v;
        }
    }

    // Padded tap tables: s_w*[d+15] = w[d] for 0<=d<=16, else 0.
    if (tid < 48) {
        bool in_rng = (tid >= 15) && (tid < 32);
        s_wx[tid] = in_rng ? taps[tid - 15]      : 0.f;
        s_wy[tid] = in_rng ? taps[KS + tid - 15] : 0.f;
    }
    __syncthreads();        // publishes TDM/scalar tile + taps to all waves

    // Loop-invariant Toeplitz fragments (layout: lanes 0-15 K={4k,4k+1},
    // lanes 16-31 K={4k+2,4k+3}).
    float bh0[8], bh1[8], av0[8], av1[8];
#pragma unroll
    for (int kk = 0; kk < 8; ++kk) {
        int kb = 4 * kk + (hs << 1);
        bh0[kk] = s_wx[kb     - cc + 15];   // B_h[k][c] = wx[k-c]
        bh1[kk] = s_wx[kb + 1 - cc + 15];
        av0[kk] = s_wy[kb     - cc + 15];   // A_v[r][k] = wy[k-r]
        av1[kk] = s_wy[kb + 1 - cc + 15];
    }

    // ---- pass 1: horizontal conv on all 80 halo rows -> s_mid ----------
    for (int t = wv; t < 20; t += 8) {      // 5 row-groups x 4 col-groups
        int r0 = (t >> 2) * 16, c0 = (t & 3) * 16;
        v8f acc = {};
#pragma unroll
        for (int kk = 0; kk < 8; ++kk) {
            int col = c0 + 4 * kk + (hs << 1);      // even mod 16: the pair
            const float* p = &s_in[tileOff(r0 + cc, col)];  // never pads
            v2f a; a.x = p[0];      a.y = p[1];
            v2f b; b.x = bh0[kk];   b.y = bh1[kk];
            acc = __builtin_amdgcn_wmma_f32_16x16x4_f32(
                      false, a, false, b, (short)0, acc, false, false);
        }
#pragma unroll
        for (int i = 0; i < 8; ++i)
            s_mid[(r0 + i + (hs << 3)) * MSTR + c0 + cc] = acc[i];
    }
    __syncthreads();

    // ---- pass 2: vertical conv + fused dot with flux -------------------
    float partial = 0.f;
    for (int t = wv; t < 16; t += 8) {      // 4 x 4 output tiles
        int r0 = (t >> 2) * 16, c0 = (t & 3) * 16;
        v8f acc = {};
#pragma unroll
        for (int kk = 0; kk < 8; ++kk) {
            int kb = r0 + 4 * kk + (hs << 1);
            v2f b; b.x = s_mid[kb * MSTR + c0 + cc];
                   b.y = s_mid[(kb + 1) * MSTR + c0 + cc];
            v2f a; a.x = av0[kk]; a.y = av1[kk];
            acc = __builtin_amdgcn_wmma_f32_16x16x4_f32(
                      false, a, false, b, (short)0, acc, false, false);
        }
#pragma unroll
        for (int i = 0; i < 8; ++i) {       // smooth * flux, local reduce
            int r = r0 + i + (hs << 3), c = c0 + cc;
            partial += acc[i] * s_in[tileOff(r + HALO, c + HALO)];
        }
    }

    // ---- block reduction -> one global float atomic --------------------
#pragma unroll
    for (int off = 16; off > 0; off >>= 1)
        partial += __shfl_xor(partial, off, 32);
    if (lane == 0) s_red[wv] = partial;
    __syncthreads();
    if (tid == 0) {
        float s = 0.f;
#pragma unroll
        for (int i = 0; i < 8; ++i) s += s_red[i];
        atomicAdd(out, -s);                 // global_atomic_add_f32
    }
}

// ---------------------------------------------------------------------------
extern "C" void kernel_launch(void* const* d_in, const int* in_sizes, int n_in,
                              void* d_out, int out_size, void* d_ws, size_t ws_size,
                              hipStream_t stream) {
    const float* flux = (const float*)d_in[0];   // [B,1,2048,2048] f32
    const float* k2d  = (const float*)d_in[1];   // [1,1,17,17]     f32
    float* out  = (float*)d_out;                 // scalar
    float* taps = (float*)d_ws;                  // 34 floats of scratch

    int batches = in_sizes[0] / (IMG_H * IMG_W);
    if (batches < 1) batches = 1;

    sp_prep_kernel<<<dim3(1), dim3(64), 0, stream>>>(k2d, taps, out);

    dim3 grid(IMG_W / TS, IMG_H / TS, batches);
    sp_main_kernel<<<grid, dim3(NTHREADS), 0, stream>>>(flux, taps, out);
}